// KEDGN_59253368815849
// MI455X (gfx1250) — compile-verified
//
#include <hip/hip_runtime.h>
#include <math.h>

// ---------------------------------------------------------------------------
// KEDGN recurrence on MI455X (gfx1250), f32 WMMA (V_WMMA_F32_16X16X4_F32).
// 64 persistent workgroups, grid barrier; phase A = per-batch comb GEMM,
// phase B = per-variable gate GEMMs with LDS-resident 129x64 weights.
// ---------------------------------------------------------------------------

typedef __attribute__((ext_vector_type(2))) float v2f;
typedef __attribute__((ext_vector_type(8))) float v8f;

#define B_   64
#define T_   128
#define V_   64
#define D_   64
#define QD_  5
#define NE_  8
#define PD_  768
#define HF_  128
#define IS_  129   // 2*D + 1
#define KP_  132   // IS padded to multiple of 4
#define SZB_ 144   // phase-A Z row stride (absorbs N-tile overrun to col 143)
#define SCA_ 66    // cur_adj row stride (bank-conflict-free A fragments)
#define NW_  8     // waves per block (256 threads / 32)
#define ALPHA_ 0.5f

__device__ __forceinline__ v8f vzero8() {
  v8f z = {0.f,0.f,0.f,0.f,0.f,0.f,0.f,0.f};
  return z;
}

__device__ __forceinline__ v8f wmma4(v2f a, v2f b, v8f c) {
  // D = A(16x4,f32) x B(4x16,f32) + C(16x16,f32)
  return __builtin_amdgcn_wmma_f32_16x16x4_f32(false, a, false, b, (short)0, c,
                                               false, false);
}

__device__ __forceinline__ float sigmoidf_(float x) {
  return 1.0f / (1.0f + __expf(-x));
}

// Device-wide sense/generation barrier for the persistent kernel.
__device__ __forceinline__ void gbar(unsigned* cnt, unsigned* gen, unsigned nb) {
  __syncthreads();
  if (threadIdx.x == 0) {
    __threadfence();
    unsigned g = __hip_atomic_load(gen, __ATOMIC_ACQUIRE, __HIP_MEMORY_SCOPE_AGENT);
    unsigned v = __hip_atomic_fetch_add(cnt, 1u, __ATOMIC_ACQ_REL, __HIP_MEMORY_SCOPE_AGENT);
    if (v == nb - 1u) {
      __hip_atomic_store(cnt, 0u, __ATOMIC_RELAXED, __HIP_MEMORY_SCOPE_AGENT);
      __hip_atomic_fetch_add(gen, 1u, __ATOMIC_ACQ_REL, __HIP_MEMORY_SCOPE_AGENT);
    } else {
      while (__hip_atomic_load(gen, __ATOMIC_ACQUIRE, __HIP_MEMORY_SCOPE_AGENT) == g) {
        __builtin_amdgcn_s_sleep(4);
      }
    }
  }
  __syncthreads();
}

// --------------------------- setup kernels ---------------------------------

// qv = relu(plm@Wf1+bf1)@Wf2+bf2 ; ng = relu(plm@Wg1+bg1)@Wg2+bg2
__global__ void k_setup1(const float* __restrict__ plm,
                         const float* __restrict__ Wf1, const float* __restrict__ bf1,
                         const float* __restrict__ Wf2, const float* __restrict__ bf2,
                         const float* __restrict__ Wg1, const float* __restrict__ bg1,
                         const float* __restrict__ Wg2, const float* __restrict__ bg2,
                         float* __restrict__ qv, float* __restrict__ ng) {
  __shared__ float hf[HF_];
  __shared__ float hg[HF_];
  int v = blockIdx.x, tid = threadIdx.x;
  float accf = bf1[tid], accg = bg1[tid];
  for (int k = 0; k < PD_; ++k) {
    float p = plm[v * PD_ + k];
    accf += p * Wf1[k * HF_ + tid];
    accg += p * Wg1[k * HF_ + tid];
  }
  hf[tid] = fmaxf(accf, 0.f);
  hg[tid] = fmaxf(accg, 0.f);
  __syncthreads();
  if (tid < QD_) {
    float a = bf2[tid];
    for (int j = 0; j < HF_; ++j) a += hf[j] * Wf2[j * QD_ + tid];
    qv[v * QD_ + tid] = a;
  }
  if (tid < NE_) {
    float a = bg2[tid];
    for (int j = 0; j < HF_; ++j) a += hg[j] * Wg2[j * NE_ + tid];
    ng[v * NE_ + tid] = a;
  }
}

// adj = softmax(ngn @ ngn^T), ngn = row-normalized ng
__global__ void k_setup2(const float* __restrict__ ng, float* __restrict__ adj) {
  __shared__ float s[V_];
  int v = blockIdx.x, u = threadIdx.x;
  float nv = 0.f, nu = 0.f, dp = 0.f;
  for (int k = 0; k < NE_; ++k) {
    float a = ng[v * NE_ + k], b = ng[u * NE_ + k];
    nv += a * a; nu += b * b; dp += a * b;
  }
  float den = fmaxf(sqrtf(nv), 1e-12f) * fmaxf(sqrtf(nu), 1e-12f);
  s[u] = dp / den;
  __syncthreads();
  float mx = -1e30f;
  for (int k = 0; k < V_; ++k) mx = fmaxf(mx, s[k]);
  float e = __expf(s[u] - mx);
  __syncthreads();
  s[u] = e;
  __syncthreads();
  float sum = 0.f;
  for (int k = 0; k < V_; ++k) sum += s[k];
  adj[v * V_ + u] = e / sum;
}

// per-variable gate weights:  gW*[v] = sum_q qv[v,q] * W_*[q]  (129x64 each)
__global__ void k_setup3(const float* __restrict__ qv,
                         const float* __restrict__ W_r, const float* __restrict__ b_r,
                         const float* __restrict__ W_u, const float* __restrict__ b_u,
                         const float* __restrict__ W_c, const float* __restrict__ b_c,
                         float* __restrict__ gWr, float* __restrict__ gbr,
                         float* __restrict__ gWu, float* __restrict__ gbu,
                         float* __restrict__ gWc, float* __restrict__ gbc) {
  int v = blockIdx.x;
  float q[QD_];
  for (int i = 0; i < QD_; ++i) q[i] = qv[v * QD_ + i];
  const int WSZ = IS_ * D_;
  for (int idx = threadIdx.x; idx < WSZ; idx += blockDim.x) {
    float ar = 0.f, au = 0.f, ac = 0.f;
    for (int i = 0; i < QD_; ++i) {
      ar += q[i] * W_r[i * WSZ + idx];
      au += q[i] * W_u[i * WSZ + idx];
      ac += q[i] * W_c[i * WSZ + idx];
    }
    gWr[v * WSZ + idx] = ar;
    gWu[v * WSZ + idx] = au;
    gWc[v * WSZ + idx] = ac;
  }
  for (int d = threadIdx.x; d < D_; d += blockDim.x) {
    float ar = 0.f, au = 0.f, ac = 0.f;
    for (int i = 0; i < QD_; ++i) {
      ar += q[i] * b_r[i * D_ + d];
      au += q[i] * b_u[i * D_ + d];
      ac += q[i] * b_c[i * D_ + d];
    }
    gbr[v * D_ + d] = ar;
    gbu[v * D_ + d] = au;
    gbc[v * D_ + d] = ac;
  }
}

// var_total[b,v] = sum_t mask[b,t,v]; also reset grid barrier.
__global__ void k_setup4(const float* __restrict__ msk, float* __restrict__ vt,
                         unsigned* __restrict__ bar) {
  int b = blockIdx.x, v = threadIdx.x;
  float s = 0.f;
  for (int t = 0; t < T_; ++t) s += msk[(b * T_ + t) * V_ + v];
  vt[b * V_ + v] = s;
  if (b == 0 && v == 0) { bar[0] = 0u; bar[1] = 0u; }
}

// --------------------------- main persistent kernel -------------------------

__global__ __launch_bounds__(256)
void k_main(const float* __restrict__ obs, const float* __restrict__ msk,
            const float* __restrict__ avg, const int* __restrict__ len,
            const float* __restrict__ rw,  const float* __restrict__ adj,
            const float* __restrict__ gWr, const float* __restrict__ gWu,
            const float* __restrict__ gWc,
            const float* __restrict__ gbr, const float* __restrict__ gbu,
            const float* __restrict__ gbc,
            const float* __restrict__ vt,
            float* __restrict__ hG, float* __restrict__ combG,
            unsigned* bar, float* __restrict__ outG) {
  __shared__ float sW[3 * KP_ * D_];            // 3 x 132x64 gate weights (~99KB)
  __shared__ float sBuf[SCA_ * V_ + SZB_ * V_]; // phaseA: curadj+Z / phaseB: comb/Zc
  __shared__ float sRar[V_];
  __shared__ float sMask[V_];

  const int bid = blockIdx.x;     // = batch b in phase A, variable v in phase B
  const int tid = threadIdx.x;
  const int wv = tid >> 5, lane = tid & 31;
  const int hlf = lane >> 4, l16 = lane & 15;

  float* sWr = sW;
  float* sWu = sW + KP_ * D_;
  float* sWc = sW + 2 * KP_ * D_;

  // Load this variable's gate weights once (resident for all 128 steps);
  // pad rows 129..131 with zeros so K-padding contributes nothing.
  for (int idx = tid; idx < KP_ * D_; idx += 256) {
    int row = idx >> 6;
    float wr = 0.f, wu = 0.f, wc = 0.f;
    if (row < IS_) {
      wr = gWr[bid * IS_ * D_ + idx];
      wu = gWu[bid * IS_ * D_ + idx];
      wc = gWc[bid * IS_ * D_ + idx];
    }
    sWr[idx] = wr; sWu[idx] = wu; sWc[idx] = wc;
  }
  // h0 = 0 for batch bid
  for (int idx = tid; idx < V_ * D_; idx += 256) hG[bid * V_ * D_ + idx] = 0.f;

  gbar(bar, bar + 1, gridDim.x);

  for (int t = 0; t < T_; ++t) {
    // ---------------- Phase A: b = bid : comb_b = cur_adj_b @ [x|rar|h]_b ----
    float* curadj = sBuf;              // [64][66]
    float* Zb     = sBuf + V_ * SCA_;  // [64][144]
    if (tid < V_) {
      float a  = avg[(bid * T_ + t) * V_ + tid];
      float vv = vt[bid * V_ + tid];
      sRar[tid]  = ALPHA_ * tanhf(a / (vv + 1.0f));
      sMask[tid] = msk[(bid * T_ + t) * V_ + tid];
    }
    __syncthreads();
    for (int idx = tid; idx < V_ * D_; idx += 256) {
      int vv = idx >> 6, d = idx & 63;
      Zb[vv * SZB_ + d]      = obs[((bid * T_ + t) * V_ + vv) * D_ + d];
      Zb[vv * SZB_ + 65 + d] = hG[(bid * V_ + vv) * D_ + d];
    }
    if (tid < V_) Zb[tid * SZB_ + 64] = sRar[tid];
    for (int idx = tid; idx < V_ * V_; idx += 256) {
      int vv = idx >> 6, u = idx & 63;
      float val = 1.0f;
      if (vv != u) {
        float rs = -rw[idx] * fabsf(sRar[vv] - sRar[u]);
        val = adj[idx] * (1.0f + rs) * sMask[vv] * sMask[u];
      }
      curadj[vv * SCA_ + u] = val;
    }
    __syncthreads();
    // GEMM 64x64 @ 64x129 via WMMA f32 16x16x4 ; 4x9 output tiles / 8 waves.
    for (int tj = wv; tj < 36; tj += NW_) {
      int m0 = (tj / 9) * 16, n0 = (tj % 9) * 16;
      v8f acc = vzero8();
      for (int k0 = 0; k0 < V_; k0 += 4) {
        const float* ap = curadj + (m0 + l16) * SCA_ + k0 + 2 * hlf;
        v2f a; a.x = ap[0]; a.y = ap[1];
        const float* bp = Zb + (k0 + 2 * hlf) * SZB_ + n0 + l16;
        v2f bb; bb.x = bp[0]; bb.y = bp[SZB_];
        acc = wmma4(a, bb, acc);
      }
      int col = n0 + l16;
      if (col < IS_) {
        int rbase = m0 + 8 * hlf;
        #pragma unroll
        for (int j = 0; j < 8; ++j)
          combG[(bid * V_ + rbase + j) * KP_ + col] = acc[j];
      }
    }
    gbar(bar, bar + 1, gridDim.x);

    // ---------------- Phase B: v = bid : gate GEMMs over batch --------------
    float* combA = sBuf;  // [64][132], K padded with zeros
    for (int idx = tid; idx < B_ * KP_; idx += 256) {
      int brow = idx / KP_, col = idx - brow * KP_;
      combA[idx] = (col < IS_) ? combG[(brow * V_ + bid) * KP_ + col] : 0.f;
    }
    __syncthreads();

    v8f uacc[2], hmid[2];
    #pragma unroll
    for (int s = 0; s < 2; ++s) {
      int tj = wv + s * NW_;
      int m0 = (tj >> 2) * 16, n0 = (tj & 3) * 16;
      v8f au = vzero8(), ar = vzero8();
      for (int k0 = 0; k0 < KP_; k0 += 4) {
        const float* ap = combA + (m0 + l16) * KP_ + k0 + 2 * hlf;
        v2f a; a.x = ap[0]; a.y = ap[1];
        const float* pu = sWu + (k0 + 2 * hlf) * D_ + n0 + l16;
        v2f bu_f; bu_f.x = pu[0]; bu_f.y = pu[D_];
        au = wmma4(a, bu_f, au);
        const float* pr = sWr + (k0 + 2 * hlf) * D_ + n0 + l16;
        v2f br_f; br_f.x = pr[0]; br_f.y = pr[D_];
        ar = wmma4(a, br_f, ar);
      }
      int dcol = n0 + l16;
      float bub = gbu[bid * D_ + dcol], brb = gbr[bid * D_ + dcol];
      #pragma unroll
      for (int j = 0; j < 8; ++j) {
        int brow = m0 + 8 * hlf + j;
        float m    = msk[(brow * T_ + t) * V_ + bid];
        float hold = hG[(brow * V_ + bid) * D_ + dcol];
        float rr = sigmoidf_(ar[j] + brb);
        float uu = sigmoidf_(au[j] + bub);
        bool on = m > 0.f;
        uacc[s][j] = on ? uu : 0.f;        // u folded with mask
        hmid[s][j] = on ? rr * hold : hold;
      }
    }
    __syncthreads();  // everyone done reading combA

    float* Zc = combA;  // reuse as [x | rar | h_mid], pads already zero
    #pragma unroll
    for (int s = 0; s < 2; ++s) {
      int tj = wv + s * NW_;
      int m0 = (tj >> 2) * 16, n0 = (tj & 3) * 16, dcol = n0 + l16;
      #pragma unroll
      for (int j = 0; j < 8; ++j)
        Zc[(m0 + 8 * hlf + j) * KP_ + 65 + dcol] = hmid[s][j];
    }
    for (int idx = tid; idx < B_ * D_; idx += 256) {
      int brow = idx >> 6, d = idx & 63;
      Zc[brow * KP_ + d] = obs[((brow * T_ + t) * V_ + bid) * D_ + d];
    }
    if (tid < B_) {
      float a  = avg[(tid * T_ + t) * V_ + bid];
      float vv = vt[tid * V_ + bid];
      Zc[tid * KP_ + 64] = ALPHA_ * tanhf(a / (vv + 1.0f));
    }
    __syncthreads();

    #pragma unroll
    for (int s = 0; s < 2; ++s) {
      int tj = wv + s * NW_;
      int m0 = (tj >> 2) * 16, n0 = (tj & 3) * 16;
      v8f ac = vzero8();
      for (int k0 = 0; k0 < KP_; k0 += 4) {
        const float* ap = Zc + (m0 + l16) * KP_ + k0 + 2 * hlf;
        v2f a; a.x = ap[0]; a.y = ap[1];
        const float* pc = sWc + (k0 + 2 * hlf) * D_ + n0 + l16;
        v2f bc_f; bc_f.x = pc[0]; bc_f.y = pc[D_];
        ac = wmma4(a, bc_f, ac);
      }
      int dcol = n0 + l16;
      float bcb = gbc[bid * D_ + dcol];
      #pragma unroll
      for (int j = 0; j < 8; ++j) {
        int brow = m0 + 8 * hlf + j;
        float cand = tanhf(ac[j] + bcb);
        float uu = uacc[s][j], hm = hmid[s][j];
        float hn = (1.f - uu) * hm + uu * cand;
        hG[(brow * V_ + bid) * D_ + dcol] = hn;
        if (t == len[brow] - 1) outG[(brow * V_ + bid) * D_ + dcol] = hn;
      }
    }
    gbar(bar, bar + 1, gridDim.x);
  }
}

// --------------------------- host launcher ----------------------------------

extern "C" void kernel_launch(void* const* d_in, const int* in_sizes, int n_in,
                              void* d_out, int out_size, void* d_ws, size_t ws_size,
                              hipStream_t stream) {
  const float* obs = (const float*)d_in[0];
  const float* msk = (const float*)d_in[1];
  const float* avg = (const float*)d_in[2];
  const int*   len = (const int*)d_in[3];
  const float* plm = (const float*)d_in[4];
  const float* Wf1 = (const float*)d_in[5];
  const float* bf1 = (const float*)d_in[6];
  const float* Wf2 = (const float*)d_in[7];
  const float* bf2 = (const float*)d_in[8];
  const float* Wg1 = (const float*)d_in[9];
  const float* bg1 = (const float*)d_in[10];
  const float* Wg2 = (const float*)d_in[11];
  const float* bg2 = (const float*)d_in[12];
  const float* rw  = (const float*)d_in[13];
  const float* W_u = (const float*)d_in[14];
  const float* b_u = (const float*)d_in[15];
  const float* W_r = (const float*)d_in[16];
  const float* b_r = (const float*)d_in[17];
  const float* W_c = (const float*)d_in[18];
  const float* b_c = (const float*)d_in[19];
  (void)in_sizes; (void)n_in; (void)out_size; (void)ws_size;

  // workspace carve-up (~9.7 MB of f32)
  float* ws   = (float*)d_ws;
  float* qv   = ws;  ws += V_ * QD_;
  float* ng   = ws;  ws += V_ * NE_;
  float* adjW = ws;  ws += V_ * V_;
  float* gWr  = ws;  ws += V_ * IS_ * D_;
  float* gWu  = ws;  ws += V_ * IS_ * D_;
  float* gWc  = ws;  ws += V_ * IS_ * D_;
  float* gbr  = ws;  ws += V_ * D_;
  float* gbu  = ws;  ws += V_ * D_;
  float* gbc  = ws;  ws += V_ * D_;
  float* vt   = ws;  ws += B_ * V_;
  float* hG   = ws;  ws += B_ * V_ * D_;
  float* comb = ws;  ws += B_ * V_ * KP_;
  unsigned* bar = (unsigned*)ws;

  k_setup1<<<V_, HF_, 0, stream>>>(plm, Wf1, bf1, Wf2, bf2, Wg1, bg1, Wg2, bg2,
                                   qv, ng);
  k_setup2<<<V_, V_, 0, stream>>>(ng, adjW);
  k_setup3<<<V_, 256, 0, stream>>>(qv, W_r, b_r, W_u, b_u, W_c, b_c,
                                   gWr, gbr, gWu, gbu, gWc, gbc);
  k_setup4<<<B_, V_, 0, stream>>>(msk, vt, bar);
  k_main<<<B_, 256, 0, stream>>>(obs, msk, avg, len, rw, adjW,
                                 gWr, gWu, gWc, gbr, gbu, gbc, vt,
                                 hG, comb, bar, (float*)d_out);
}